// PostProcess_54494545052409
// MI455X (gfx1250) — compile-verified
//
#include <hip/hip_runtime.h>
#include <hip/hip_bf16.h>
#include <cstdint>

// ---------------------------------------------------------------------------
// DETR post-process for MI455X (gfx1250, wave32).
// Phases:
//   K1: softmax-max score + argmax label + box convert/scale (1 wave per row)
//   K2: per-batch LDS bitonic sort (2048 keys) -> sorted outputs + offset boxes
//   K3: tiled pairwise-IoU suppression bitmask; union base via
//       V_WMMA_F32_16X16X4_F32 outer-product-sum; row tiles staged in LDS via
//       async global->LDS loads
//   K4: sequential 2048-step keep scan per batch (64 threads, bitset in LDS)
// ---------------------------------------------------------------------------

#define NMS_TH     0.7f
#define SCORE_TH   0.05f
#define LAB_OFF    100000.0f
#define NB         16
#define NQ         2048
#define NC         256

#if __has_builtin(__builtin_amdgcn_wmma_f32_16x16x4_f32)
#define USE_WMMA_F32X4 1
#else
#pragma message("wmma_f32_16x16x4_f32 builtin NOT available - VALU fallback")
#endif

#if __has_builtin(__builtin_amdgcn_global_load_async_to_lds_b32)
#define USE_ASYNC_LDS 1
#else
#pragma message("global_load_async_to_lds_b32 builtin NOT available - plain LDS store fallback")
#endif

typedef __attribute__((ext_vector_type(2))) float v2f;
typedef __attribute__((ext_vector_type(8))) float v8f;

typedef __attribute__((address_space(1))) int gas_int;   // global (AS1)
typedef __attribute__((address_space(3))) int las_int;   // LDS (AS3)

__device__ __forceinline__ void g2lds_b32(const float* g, float* l) {
#if defined(USE_ASYNC_LDS)
  float* gnc = const_cast<float*>(g);
  __builtin_amdgcn_global_load_async_to_lds_b32(
      (gas_int*)gnc, (las_int*)l, 0, 0);
#else
  *l = *g;
#endif
}

// ---------------------------------------------------------------------------
// K1: one wave32 per (b,q) row: 8 classes per lane.
// score = exp(max_{c<255} x - max_all x) / sum exp(x - max_all)
// ---------------------------------------------------------------------------
__global__ void k1_score_label_box(const float* __restrict__ logits,
                                   const float* __restrict__ boxes,
                                   const int*   __restrict__ tsizes,
                                   float* __restrict__ wBoxes,
                                   float* __restrict__ wScores,
                                   int*   __restrict__ wLabels) {
  const int wave = threadIdx.x >> 5;
  const int lane = threadIdx.x & 31;
  const int row  = blockIdx.x * 8 + wave;           // 0 .. NB*NQ-1

  const float* rp = logits + (size_t)row * NC + lane * 8;
  float4 a = ((const float4*)rp)[0];
  float4 b = ((const float4*)rp)[1];

  float m = fmaxf(fmaxf(fmaxf(a.x, a.y), fmaxf(a.z, a.w)),
                  fmaxf(fmaxf(b.x, b.y), fmaxf(b.z, b.w)));
  #pragma unroll
  for (int off = 16; off; off >>= 1) m = fmaxf(m, __shfl_xor(m, off, 32));

  float s = __expf(a.x - m) + __expf(a.y - m) + __expf(a.z - m) + __expf(a.w - m) +
            __expf(b.x - m) + __expf(b.y - m) + __expf(b.z - m) + __expf(b.w - m);
  #pragma unroll
  for (int off = 16; off; off >>= 1) s += __shfl_xor(s, off, 32);

  // argmax over first NC-1 classes (class 255 lives in lane 31, elem 7)
  float v[8] = {a.x, a.y, a.z, a.w, b.x, b.y, b.z, b.w};
  float bm = -INFINITY; int bi = 0;
  #pragma unroll
  for (int e = 0; e < 8; ++e) {
    int c = lane * 8 + e;
    if (c < NC - 1 && v[e] > bm) { bm = v[e]; bi = c; }
  }
  #pragma unroll
  for (int off = 16; off; off >>= 1) {
    float om = __shfl_xor(bm, off, 32);
    int   oi = __shfl_xor(bi, off, 32);
    if (om > bm || (om == bm && oi < bi)) { bm = om; bi = oi; }
  }

  if (lane == 0) {
    wScores[row] = __expf(bm - m) / s;
    wLabels[row] = bi;
    const int bb = row / NQ;
    const float ih = (float)tsizes[bb * 2 + 0];
    const float iw = (float)tsizes[bb * 2 + 1];
    float4 bx = ((const float4*)boxes)[row];       // cx, cy, w, h
    float4 o;
    o.x = (bx.x - 0.5f * bx.z) * iw;
    o.y = (bx.y - 0.5f * bx.w) * ih;
    o.z = (bx.x + 0.5f * bx.z) * iw;
    o.w = (bx.y + 0.5f * bx.w) * ih;
    ((float4*)wBoxes)[row] = o;
  }
}

// ---------------------------------------------------------------------------
// K2: per-batch descending bitonic sort of 2048 (key,idx) in LDS, then gather
// sorted outputs + label-offset boxes, areas, validity.
// ---------------------------------------------------------------------------
__global__ void k2_sort_gather(const float* __restrict__ wScores,
                               const int*   __restrict__ wLabels,
                               const float* __restrict__ wBoxes,
                               float* __restrict__ oBoxes,
                               float* __restrict__ oScores,
                               float* __restrict__ oLabels,
                               float* __restrict__ wBoxOff,
                               float* __restrict__ wArea,
                               unsigned* __restrict__ wValid) {
  __shared__ float sk[NQ];
  __shared__ int   sv[NQ];
  const int b = blockIdx.x, t = threadIdx.x, base = b * NQ;

  for (int i = t; i < NQ; i += 1024) {
    float sc = wScores[base + i];
    sk[i] = (sc >= SCORE_TH) ? sc : -INFINITY;     // sort key (reference: where(valid, s, -inf))
    sv[i] = i;
  }
  for (int k = 2; k <= NQ; k <<= 1) {
    for (int j = k >> 1; j > 0; j >>= 1) {
      __syncthreads();
      for (int i = t; i < NQ; i += 1024) {
        int ixj = i ^ j;
        if (ixj > i) {
          float ki = sk[i], kj = sk[ixj];
          bool sw = ((i & k) == 0) ? (ki < kj) : (ki > kj);   // descending
          if (sw) {
            sk[i] = kj; sk[ixj] = ki;
            int tv = sv[i]; sv[i] = sv[ixj]; sv[ixj] = tv;
          }
        }
      }
    }
  }
  __syncthreads();
  for (int p = t; p < NQ; p += 1024) {
    int o = sv[p];
    float sc  = wScores[base + o];
    int   lab = wLabels[base + o];
    float4 bx = ((const float4*)wBoxes)[base + o];
    ((float4*)oBoxes)[base + p] = bx;
    oScores[base + p] = sc;
    oLabels[base + p] = (float)lab;
    float off = (float)lab * LAB_OFF;
    float4 bo = make_float4(bx.x + off, bx.y + off, bx.z + off, bx.w + off);
    ((float4*)wBoxOff)[base + p] = bo;
    wArea[base + p] = fmaxf(bo.z - bo.x, 0.f) * fmaxf(bo.w - bo.y, 0.f);
    wValid[base + p] = (sc >= SCORE_TH) ? 1u : 0u;
  }
}

// ---------------------------------------------------------------------------
// K3: one block per (batch, 16-row tile); each wave32 covers 16x32 IoU tiles.
// union base (area_i + area_j) via one WMMA f32 16x16x4 per 16x16 subtile.
// ---------------------------------------------------------------------------
__global__ void k3_iou_mask(const float* __restrict__ boxoff,
                            const float* __restrict__ area,
                            unsigned* __restrict__ mask) {
  __shared__ float sRowBox[64];    // 16 boxes * 4 coords
  __shared__ float sRowArea[16];
  const int b    = blockIdx.x >> 7;
  const int ti   = blockIdx.x & 127;
  const int base = b * NQ;
  const int ig0  = ti * 16;
  const int t    = threadIdx.x;

  if (t < 64)       g2lds_b32(&boxoff[(size_t)(base + ig0) * 4 + t], &sRowBox[t]);
  else if (t < 80)  g2lds_b32(&area[base + ig0 + (t - 64)], &sRowArea[t - 64]);
#if defined(USE_ASYNC_LDS)
  __builtin_amdgcn_s_wait_asynccnt(0);
#endif
  __syncthreads();

  const int wave = t >> 5, lane = t & 31;
  const int N = lane & 15;

  v2f A;
  A.x = (lane < 16) ? sRowArea[N] : 0.f;   // K=0 column: area of row M=lane
  A.y = (lane < 16) ? 1.f : 0.f;           // K=1 column: ones

  for (int it = 0; it < 8; ++it) {
    const int cp  = wave * 8 + it;         // 32-column group 0..63
    const int jg0 = cp * 32;

    float4 cb0 = ((const float4*)boxoff)[base + jg0 + N];
    float4 cb1 = ((const float4*)boxoff)[base + jg0 + 16 + N];
    float  ca0 = area[base + jg0 + N];
    float  ca1 = area[base + jg0 + 16 + N];

#if defined(USE_WMMA_F32X4)
    v2f B0, B1;
    B0.x = (lane < 16) ? 1.f : ca0;  B0.y = 0.f;   // row K=0: ones, K=1: area_col
    B1.x = (lane < 16) ? 1.f : ca1;  B1.y = 0.f;
    v8f Z = {};
    // D[M][N] = area_row[M]*1 + 1*area_col[N]
    v8f U0 = __builtin_amdgcn_wmma_f32_16x16x4_f32(false, A, false, B0, (short)0, Z, false, false);
    v8f U1 = __builtin_amdgcn_wmma_f32_16x16x4_f32(false, A, false, B1, (short)0, Z, false, false);
#endif

    #pragma unroll
    for (int r = 0; r < 8; ++r) {
      const int M = r + ((lane >= 16) ? 8 : 0);
      const float rx0 = sRowBox[M * 4 + 0], ry0 = sRowBox[M * 4 + 1];
      const float rx1 = sRowBox[M * 4 + 2], ry1 = sRowBox[M * 4 + 3];

      float u0, u1;
#if defined(USE_WMMA_F32X4)
      u0 = U0[r]; u1 = U1[r];
#else
      u0 = sRowArea[M] + ca0; u1 = sRowArea[M] + ca1;
#endif
      // tile 0: cols jg0 .. jg0+15
      float in0 = fmaxf(fminf(rx1, cb0.z) - fmaxf(rx0, cb0.x), 0.f) *
                  fmaxf(fminf(ry1, cb0.w) - fmaxf(ry0, cb0.y), 0.f);
      bool p0 = (in0 > NMS_TH * fmaxf(u0 - in0, 1e-6f)) && ((jg0 + N) > (ig0 + M));
      // tile 1: cols jg0+16 .. jg0+31
      float in1 = fmaxf(fminf(rx1, cb1.z) - fmaxf(rx0, cb1.x), 0.f) *
                  fmaxf(fminf(ry1, cb1.w) - fmaxf(ry0, cb1.y), 0.f);
      bool p1 = (in1 > NMS_TH * fmaxf(u1 - in1, 1e-6f)) && ((jg0 + 16 + N) > (ig0 + M));

      unsigned b0 = (unsigned)__ballot(p0);
      unsigned b1 = (unsigned)__ballot(p1);
      if (lane == 0) {
        unsigned lowRow  = (b0 & 0xFFFFu) | ((b1 & 0xFFFFu) << 16);
        unsigned highRow = (b0 >> 16)     | (b1 & 0xFFFF0000u);
        mask[(size_t)(base + ig0 + r)     * 64 + cp] = lowRow;
        mask[(size_t)(base + ig0 + r + 8) * 64 + cp] = highRow;
      }
    }
  }
}

// ---------------------------------------------------------------------------
// K4: per-batch sequential suppression scan (torchvision semantics == the
// reference lax.scan). 64 threads hold the 2048-bit removed-set in LDS.
// ---------------------------------------------------------------------------
__global__ void k4_nms_scan(const unsigned* __restrict__ valid,
                            const unsigned* __restrict__ mask,
                            float* __restrict__ oKeep) {
  __shared__ unsigned remv[64];
  const int b = blockIdx.x, t = threadIdx.x, base = b * NQ;
  remv[t] = 0u;
  __syncthreads();
  for (int i = 0; i < NQ; ++i) {
    const unsigned removed = (remv[i >> 5] >> (i & 31)) & 1u;
    const unsigned keep_i  = valid[base + i] & (removed ^ 1u);
    if (i + 1 < NQ) __builtin_prefetch(&mask[(size_t)(base + i + 1) * 64 + t], 0, 0);
    __syncthreads();                       // reads of remv done before updates
    if (keep_i) remv[t] |= mask[(size_t)(base + i) * 64 + t];
    if (t == 0) oKeep[base + i] = keep_i ? 1.0f : 0.0f;
    __syncthreads();
  }
}

// ---------------------------------------------------------------------------
extern "C" void kernel_launch(void* const* d_in, const int* in_sizes, int n_in,
                              void* d_out, int out_size, void* d_ws, size_t ws_size,
                              hipStream_t stream) {
  const float* logits = (const float*)d_in[0];   // [B,Q,C]
  const float* boxes  = (const float*)d_in[1];   // [B,Q,4]
  const int*   tsizes = (const int*)d_in[2];     // [B,2]

  float* out_boxes  = (float*)d_out;                       // [B,Q,4]
  float* out_scores = out_boxes  + NB * NQ * 4;            // [B,Q]
  float* out_labels = out_scores + NB * NQ;                // [B,Q] (as f32)
  float* out_keep   = out_labels + NB * NQ;                // [B,Q] (as f32 0/1)

  // workspace layout (all 4-byte elements, ~10 MB total)
  float*    wBoxes  = (float*)d_ws;                        // B*Q*4
  float*    wScores = wBoxes  + NB * NQ * 4;               // B*Q
  int*      wLabels = (int*)(wScores + NB * NQ);           // B*Q
  float*    wBoxOff = (float*)(wLabels + NB * NQ);         // B*Q*4
  float*    wArea   = wBoxOff + NB * NQ * 4;               // B*Q
  unsigned* wValid  = (unsigned*)(wArea + NB * NQ);        // B*Q
  unsigned* wMask   = wValid + NB * NQ;                    // B*Q*(Q/32)

  k1_score_label_box<<<NB * NQ / 8, 256, 0, stream>>>(logits, boxes, tsizes,
                                                      wBoxes, wScores, wLabels);
  k2_sort_gather<<<NB, 1024, 0, stream>>>(wScores, wLabels, wBoxes,
                                          out_boxes, out_scores, out_labels,
                                          wBoxOff, wArea, wValid);
  k3_iou_mask<<<NB * 128, 256, 0, stream>>>(wBoxOff, wArea, wMask);
  k4_nms_scan<<<NB, 64, 0, stream>>>(wValid, wMask, out_keep);
}